// AttnPhi_5428838662167
// MI455X (gfx1250) — compile-verified
//
#include <hip/hip_runtime.h>
#include <hip/hip_bf16.h>
#include <math.h>

// Problem constants (from the reference)
#define BB_   16
#define TT_   4096
#define HH_   8
#define DD_   64
#define CC_   512   // HH_*DD_

typedef __attribute__((ext_vector_type(2))) float v2f;
typedef __attribute__((ext_vector_type(8))) float v8f;

// ---------------------------------------------------------------------------
// Kernel 1: per-batch cumsum of alphas -> firing points fp[b,t], out_lens[b].
// One block (256 threads) per batch, 16 elements per thread, block scan.
// jnp.round == round-half-to-even == rintf under default rounding mode.
// ---------------------------------------------------------------------------
__global__ void cif_scan_kernel(const float* __restrict__ alphas,     // (B,T,1)
                                const unsigned char* __restrict__ mask, // (B,T) bool
                                int* __restrict__ fp,                 // (B,T)
                                int* __restrict__ olens,              // (B,)
                                float* __restrict__ out_tail,         // d_out + B*T1*C
                                int T1) {
    __shared__ float s_tot[256];
    __shared__ float s_total;
    const int b   = blockIdx.x;
    const int tid = threadIdx.x;

    float loc[16];
    float run = 0.0f;
    const float* a = alphas + (size_t)b * TT_ + tid * 16;
#pragma unroll
    for (int i = 0; i < 16; ++i) { run += a[i]; loc[i] = run; }
    s_tot[tid] = run;
    __syncthreads();

    if (tid == 0) {
        float acc = 0.0f;
        for (int j = 0; j < 256; ++j) { float t = s_tot[j]; s_tot[j] = acc; acc += t; }
        s_total = acc;
    }
    __syncthreads();

    const float base = s_tot[tid];
    const int   oL   = max(1, (int)rintf(s_total));
    const unsigned char* mb = mask + (size_t)b * TT_;
    int* fpb = fp + (size_t)b * TT_;

#pragma unroll
    for (int i = 0; i < 16; ++i) {
        const int t = tid * 16 + i;      // cs[t] = base + loc[i]; drives fp[t+1]
        if (t + 1 < TT_) {
            const float cs = base + loc[i];
            int fv = (int)rintf(cs - 0.5f);
            const bool drop = (mb[t + 1] != 0) || (fv == oL);
            fpb[t + 1] = drop ? T1 : fv;
        }
    }
    if (tid == 0) {
        fpb[0]      = (mb[0] != 0) ? T1 : 0;   // oL >= 1, so fp==oL impossible at 0
        olens[b]    = oL;
        out_tail[b] = (float)oL;               // second tuple output, as float
    }
}

// ---------------------------------------------------------------------------
// Kernel 2: rel-pos table pos[t, c] = trig((2048 - t) * div_{c/2}) * 512^-0.5
// ---------------------------------------------------------------------------
__global__ void pos_table_kernel(float* __restrict__ pos) {
    const int idx = blockIdx.x * blockDim.x + threadIdx.x;
    if (idx >= TT_ * CC_) return;
    const int t = idx / CC_;
    const int c = idx - t * CC_;
    const int i = c >> 1;
    const float div   = expf((float)(2 * i) * (-0.01798894603901598f)); // -ln(1e4)/512
    const float angle = (float)(2048 - t) * div;
    const float v     = (c & 1) ? cosf(angle) : sinf(angle);
    pos[idx] = v * 0.04419417382415922f;                                // 512^-0.5
}

// ---------------------------------------------------------------------------
// Kernel 3: scores[b,h,t] = key[b,h,t,:] . query[h,:] via V_WMMA_F32_16X16X4_F32.
// One wave per (b,h, 16-token tile). B-matrix columns all replicate query so
// every D column equals the score vector. A layout (16x4 f32): lane L holds
// (m = L&15, k = 2*(L>>4) + {0,1}); B mirrors the same K slotting.
// D layout: lane 0 regs 0..7 -> tokens m=0..7 (N=0); lane 16 -> m=8..15.
// ---------------------------------------------------------------------------
__global__ void scores_wmma_kernel(const float* __restrict__ src,   // (B,T,C)
                                   const float* __restrict__ query, // (H,D)
                                   float* __restrict__ scores) {    // (B,H,T)
    const int wave = (blockIdx.x * blockDim.x + threadIdx.x) >> 5;
    const int totalWaves = BB_ * HH_ * (TT_ / 16);
    if (wave >= totalWaves) return;                    // wave-uniform guard

    const int tile = wave & 255;                       // T/16 = 256
    const int h    = (wave >> 8) & 7;
    const int b    = wave >> 11;
    const int t0   = tile * 16;

    const int lane = threadIdx.x & 31;
    const int m    = lane & 15;
    const int kb   = (lane >> 4) << 1;                 // 0 or 2

    const float* srow = src + ((size_t)b * TT_ + (t0 + m)) * CC_ + h * DD_;
    const float* q    = query + h * DD_;

    v8f c = {};
    for (int s = 0; s < 16; ++s) {
        const int j = 4 * s + kb;
        v2f a;  a.x  = srow[j]; a.y  = srow[j + 1];
        v2f bq; bq.x = q[j];    bq.y = q[j + 1];
        c = __builtin_amdgcn_wmma_f32_16x16x4_f32(false, a, false, bq,
                                                  (short)0, c, false, false);
    }

    if ((lane & 15) == 0) {
        const int mb = (lane >> 4) * 8;
        float* so = scores + ((size_t)(b * HH_ + h)) * TT_ + t0 + mb;
#pragma unroll
        for (int v = 0; v < 8; ++v) so[v] = c[v];
    }
}

// ---------------------------------------------------------------------------
// Kernel 4: deterministic segmented softmax + collapse.
// Block (k = blockIdx.x, b = blockIdx.y): tokens of bucket k are the
// contiguous range [lb(k), lb(k+1)) in the sorted fp[b,:]. Dustbin (fp==T1)
// never matches k < T1, so dropped tokens are skipped for free. Empty buckets
// write zeros (matches x_out zero-init).
// ---------------------------------------------------------------------------
__device__ __forceinline__ int lbound(const int* __restrict__ a, int n, int key) {
    int lo = 0, hi = n;
    while (lo < hi) { int mid = (lo + hi) >> 1; if (a[mid] < key) lo = mid + 1; else hi = mid; }
    return lo;
}

__global__ void collapse_kernel(const float* __restrict__ src,    // (B,T,C)
                                const float* __restrict__ pos,    // (T,C)
                                const float* __restrict__ scores, // (B,H,T)
                                const int* __restrict__ fp,       // (B,T)
                                float* __restrict__ out,          // (B,T1,C)
                                int T1) {
    const int k = blockIdx.x;
    const int b = blockIdx.y;
    __shared__ int   s_t0, s_t1;
    __shared__ float s_invS[HH_];

    const int* fpb = fp + (size_t)b * TT_;
    if (threadIdx.x == 0) {
        s_t0 = lbound(fpb, TT_, k);
        s_t1 = lbound(fpb, TT_, k + 1);
    }
    __syncthreads();
    const int t0 = s_t0, t1 = s_t1;

    if (threadIdx.x < HH_) {
        const float* sc = scores + ((size_t)(b * HH_ + threadIdx.x)) * TT_;
        float s = 0.0f;
        for (int t = t0; t < t1; ++t) s += expf(sc[t]);
        s_invS[threadIdx.x] = (s > 0.0f) ? (1.0f / s) : 0.0f;
    }
    __syncthreads();

    float* orow = out + ((size_t)b * T1 + k) * CC_;
    for (int c = threadIdx.x; c < CC_; c += blockDim.x) {
        const int h = c >> 6;
        const float invS = s_invS[h];
        const float* sc  = scores + ((size_t)(b * HH_ + h)) * TT_;
        float acc = 0.0f;
        for (int t = t0; t < t1; ++t) {
            const float w = expf(sc[t]) * invS;
            acc += w * (src[((size_t)b * TT_ + t) * CC_ + c] + pos[(size_t)t * CC_ + c]);
        }
        orow[c] = acc;
    }
}

// ---------------------------------------------------------------------------
extern "C" void kernel_launch(void* const* d_in, const int* in_sizes, int n_in,
                              void* d_out, int out_size, void* d_ws, size_t ws_size,
                              hipStream_t stream) {
    const float*         src    = (const float*)d_in[0];
    const unsigned char* mask   = (const unsigned char*)d_in[1]; // bool (B,T)
    const float*         alphas = (const float*)d_in[2];
    const float*         query  = (const float*)d_in[3];
    float*               out    = (float*)d_out;

    // out_size = B*T1*C + B  ->  T1
    const int T1 = (out_size - BB_) / (BB_ * CC_);

    // workspace layout
    int*   fp     = (int*)d_ws;                       // B*T
    int*   olens  = fp + BB_ * TT_;                   // B (padded to 64)
    float* pos    = (float*)(olens + 64);             // T*C
    float* scores = pos + (size_t)TT_ * CC_;          // B*H*T

    float* out_tail = out + (size_t)BB_ * T1 * CC_;   // out_lens as floats

    cif_scan_kernel<<<BB_, 256, 0, stream>>>(alphas, mask, fp, olens, out_tail, T1);

    {
        const int n = TT_ * CC_;
        pos_table_kernel<<<(n + 255) / 256, 256, 0, stream>>>(pos);
    }

    {
        const int totalWaves = BB_ * HH_ * (TT_ / 16); // 32768
        const int wavesPerBlock = 4;                   // 128 threads
        scores_wmma_kernel<<<totalWaves / wavesPerBlock, wavesPerBlock * 32, 0, stream>>>(
            src, query, scores);
    }

    {
        dim3 grid(T1, BB_);
        collapse_kernel<<<grid, 256, 0, stream>>>(src, pos, scores, fp, out, T1);
    }
}